// HGCN_66614942761370
// MI455X (gfx1250) — compile-verified
//
#include <hip/hip_runtime.h>
#include <hip/hip_bf16.h>

#define EPSV      1e-7f
#define MIN_NRM   1e-15f
#define MAX_NRM   1000000.0f
#define N_NODES   10000
#define N_EDGES   320000
#define CH        128
#define MTILES    (N_NODES / 16)   // 625

typedef __attribute__((ext_vector_type(2))) float v2f;
typedef __attribute__((ext_vector_type(8))) float v8f;
typedef __attribute__((address_space(3))) float lds_float;

// ---------------- wave32 helpers ----------------
__device__ __forceinline__ float wsum(float v) {
#pragma unroll
  for (int m = 16; m >= 1; m >>= 1) v += __shfl_xor(v, m, 32);
  return v;
}
__device__ __forceinline__ float dot4(float4 a, float4 b) {
  return a.x * b.x + a.y * b.y + a.z * b.z + a.w * b.w;
}
__device__ __forceinline__ float4 sc4(float s, float4 v) {
  return make_float4(s * v.x, s * v.y, s * v.z, s * v.w);
}

// ---------------- fp32 WMMA GEMM with LDS-staged weights -------------------------
// C[M, n0:n0+NT*16] = A[M,128] @ W[:, woff:woff+128]^T (+bias)
// Block = 8 waves; each wave owns one 16-row M tile. The weight panel for this
// block's N range is staged into LDS once via ASYNCcnt-tracked
// global_load_async_to_lds_b32, then every wave streams B fragments from LDS
// (row stride 132 floats -> conflict-free ds_load_b64) and A from global,
// issuing NT back-to-back V_WMMA_F32_16X16X4_F32 per A fragment.
template <int NT>
__global__ __launch_bounds__(256) void hgcn_wmma_mm(
    const float* __restrict__ A,
    const float* __restrict__ W, int ldw, int woff,
    const float* __restrict__ bias,
    float* __restrict__ C, int ldc, int mtiles) {
  constexpr int KK = CH;                 // K = 128
  constexpr int NROW = NT * 16;          // weight rows staged per block
  constexpr int LDS_LD = KK + 4;         // pad: lanes map to distinct banks
  __shared__ float sw[NROW * LDS_LD];

  const int tid = threadIdx.x;
  const int n0 = blockIdx.y * NROW;

  // ---- async stage: sw[n*LDS_LD + k] = W[(n0+n)*ldw + woff + k]
  for (int i = tid; i < NROW * KK; i += 256) {
    const int n = i >> 7, k = i & (KK - 1);
    const int goff = ((n0 + n) * ldw + woff + k) << 2;   // byte offset from W
    const lds_float* lp = (const lds_float*)&sw[n * LDS_LD + k];
    asm volatile("global_load_async_to_lds_b32 %0, %1, %2"
                 :: "v"(lp), "v"(goff), "s"(W) : "memory");
  }
  asm volatile("s_wait_asynccnt 0" ::: "memory");
  __syncthreads();

  const int lane = tid & 31;
  const int mtile = blockIdx.x * 8 + (tid >> 5);
  if (mtile < mtiles) {
    const int lo = lane & 15, hi = lane >> 4;
    const int m0 = mtile << 4;
    const float* ap = A + (m0 + lo) * KK;          // this lane's A row
    v8f acc[NT] = {};
    for (int k = 0; k < KK; k += 4) {
      const int kk = k + (hi << 1);                // lanes 0-15: K 0,1 ; 16-31: K 2,3
      v2f a; a[0] = ap[kk]; a[1] = ap[kk + 1];
      v2f b[NT];
#pragma unroll
      for (int t = 0; t < NT; ++t) {
        const float* bp = &sw[(t * 16 + lo) * LDS_LD + kk];
        b[t][0] = bp[0]; b[t][1] = bp[1];
      }
#pragma unroll
      for (int t = 0; t < NT; ++t)
        acc[t] = __builtin_amdgcn_wmma_f32_16x16x4_f32(
            false, a, false, b[t], (short)0, acc[t], false, false);
    }
#pragma unroll
    for (int t = 0; t < NT; ++t) {
      const int col = n0 + (t << 4) + lo;
      const float bv = bias ? bias[col] : 0.0f;
      float* cp = C + (m0 + (hi << 3)) * ldc + col;
#pragma unroll
      for (int r = 0; r < 8; ++r) cp[r * ldc] = acc[t][r] + bv;
    }
  }
}

// ---------------- node kernel A: x -> t0 = logmap0(proj(expmap0([0,x]))) ---------
// t0 stored spatial-only (128 wide), matches W0[:,1:].
__global__ __launch_bounds__(256) void hgcn_embed(const float* __restrict__ x,
                                                  float* __restrict__ t0) {
  const int lane = threadIdx.x & 31;
  const int node = blockIdx.x * 8 + (threadIdx.x >> 5);
  if (node >= N_NODES) return;
  const int c = lane << 2;
  float4 v = *(const float4*)(x + node * CH + c);
  float vn = fmaxf(sqrtf(wsum(dot4(v, v))), MIN_NRM);
  float4 y = sc4(sinhf(vn) / vn, v);
  float sy = wsum(dot4(y, y));
  float x0 = sqrtf(fmaxf(1.0f + sy, EPSV));
  float f  = acoshf(fmaxf(x0, 1.0f + EPSV)) / fmaxf(sqrtf(sy), MIN_NRM);
  *(float4*)(t0 + node * CH + c) = sc4(f, y);
}

// ---------------- bias tangent: u = logmap0(proj(expmap0(proj_tan0(b)))) ---------
__global__ __launch_bounds__(32) void hgcn_bias(const float* __restrict__ b,
                                                float* __restrict__ u) {
  const int lane = threadIdx.x & 31;
  const int c = lane << 2;
  float4 v = *(const float4*)(b + c);
  if (c == 0) v.x = 0.0f;                           // proj_tan0 zeroes slot 0
  float vn = fmaxf(sqrtf(wsum(dot4(v, v))), MIN_NRM);
  float4 y = sc4(sinhf(vn) / vn, v);
  float sy = wsum(dot4(y, y));
  float x0 = sqrtf(fmaxf(1.0f + sy, EPSV));
  float f  = acoshf(fmaxf(x0, 1.0f + EPSV)) / fmaxf(sqrtf(sy), MIN_NRM);
  *(float4*)(u + c) = sc4(f, y);                    // u[0] = 0
}

// ---------------- node kernel B: mm -> t2 = logmap0(mobius_add(proj(expmap0(mm)), bias))
__global__ __launch_bounds__(256) void hgcn_mobius(const float* __restrict__ mm,
                                                   const float* __restrict__ u,
                                                   float* __restrict__ t2) {
  const int lane = threadIdx.x & 31;
  const int node = blockIdx.x * 8 + (threadIdx.x >> 5);
  if (node >= N_NODES) return;
  const int c = lane << 2;
  float4 m = *(const float4*)(mm + node * CH + c);
  if (c == 0) m.x = 0.0f;                           // expmap0 ignores comp 0
  // mv = proj(expmap0(mm))
  float vn = fmaxf(sqrtf(wsum(dot4(m, m))), MIN_NRM);
  float4 y = sc4(sinhf(vn) / vn, m);                // spatial part of mv
  float sy = wsum(dot4(y, y));
  float x0 = sqrtf(fmaxf(1.0f + sy, EPSV));
  float yn = fmaxf(sqrtf(sy), MIN_NRM);
  // ptransp0(mv, u) then proj_tan
  float4 ub = *(const float4*)(u + c);
  float alpha = wsum(dot4(y, ub)) / yn;
  float coef  = alpha * (1.0f - x0) / yn;
  float4 w = make_float4(ub.x - coef * y.x, ub.y - coef * y.y,
                         ub.z - coef * y.z, ub.w - coef * y.w);
  float v0 = wsum(dot4(y, w)) / fmaxf(x0, EPSV);
  // expmap(v = [v0, w], x = [x0, y])
  float sww = wsum(dot4(w, w));
  float nu  = sqrtf(fmaxf(sww - v0 * v0, EPSV));    // minkowski norm
  nu = fminf(nu, MAX_NRM);
  float theta = fmaxf(nu, MIN_NRM);
  float chv = coshf(theta), shv = sinhf(theta) / theta;
  float4 res = make_float4(chv * y.x + shv * w.x, chv * y.y + shv * w.y,
                           chv * y.z + shv * w.z, chv * y.w + shv * w.w);
  // proj + logmap0
  float sres = wsum(dot4(res, res));
  float rx0  = sqrtf(fmaxf(1.0f + sres, EPSV));
  float f2   = acoshf(fmaxf(rx0, 1.0f + EPSV)) / fmaxf(sqrtf(sres), MIN_NRM);
  *(float4*)(t2 + node * CH + c) = sc4(f2, res);    // slot 0 stays 0
}

// ---------------- zero workspace ----------------
__global__ __launch_bounds__(256) void hgcn_zero(float* __restrict__ p, int n4) {
  int i = blockIdx.x * 256 + threadIdx.x;
  if (i < n4) *(float4*)(p + i * 4) = make_float4(0.f, 0.f, 0.f, 0.f);
}

// ---------------- sparse aggregation: s[e0[e]] += t2[e1[e]] ----------------------
__global__ __launch_bounds__(256) void hgcn_scatter(const int* __restrict__ e0,
                                                    const int* __restrict__ e1,
                                                    const float* __restrict__ t2,
                                                    float* __restrict__ s) {
  int idx = blockIdx.x * 256 + threadIdx.x;
  int e = idx >> 5;
  if (e >= N_EDGES) return;
  int c = (idx & 31) << 2;                          // wave-uniform e, per-lane channels
  int dst = e0[e], src = e1[e];
  float4 v = *(const float4*)(t2 + src * CH + c);
  float* o = s + dst * CH + c;
  atomicAdd(o + 0, v.x); atomicAdd(o + 1, v.y);
  atomicAdd(o + 2, v.z); atomicAdd(o + 3, v.w);
}

// ---------------- node kernel D: s -> t_next -------------------------------------
// t_next = logmap0(proj(expmap0(relu(logmap0(proj(expmap0(s)))))))
__global__ __launch_bounds__(256) void hgcn_actnext(const float* __restrict__ s,
                                                    float* __restrict__ tn) {
  const int lane = threadIdx.x & 31;
  const int node = blockIdx.x * 8 + (threadIdx.x >> 5);
  if (node >= N_NODES) return;
  const int c = lane << 2;
  float4 v = *(const float4*)(s + node * CH + c);
  if (c == 0) v.x = 0.0f;
  float vn = fmaxf(sqrtf(wsum(dot4(v, v))), MIN_NRM);
  float4 y = sc4(sinhf(vn) / vn, v);
  float sy = wsum(dot4(y, y));
  float x0 = sqrtf(fmaxf(1.0f + sy, EPSV));
  float f1 = acoshf(fmaxf(x0, 1.0f + EPSV)) / fmaxf(sqrtf(sy), MIN_NRM);
  float4 t = make_float4(fmaxf(f1 * y.x, 0.f), fmaxf(f1 * y.y, 0.f),
                         fmaxf(f1 * y.z, 0.f), fmaxf(f1 * y.w, 0.f));  // relu
  float vn2 = fmaxf(sqrtf(wsum(dot4(t, t))), MIN_NRM);
  float4 y2 = sc4(sinhf(vn2) / vn2, t);
  float sy2 = wsum(dot4(y2, y2));
  float x0b = sqrtf(fmaxf(1.0f + sy2, EPSV));
  float f2  = acoshf(fmaxf(x0b, 1.0f + EPSV)) / fmaxf(sqrtf(sy2), MIN_NRM);
  *(float4*)(tn + node * CH + c) = sc4(f2, y2);
}

// ---------------- launch ----------------
extern "C" void kernel_launch(void* const* d_in, const int* in_sizes, int n_in,
                              void* d_out, int out_size, void* d_ws, size_t ws_size,
                              hipStream_t stream) {
  const float* x  = (const float*)d_in[0];
  const float* W0 = (const float*)d_in[1];
  const float* b0 = (const float*)d_in[2];
  const float* W1 = (const float*)d_in[3];
  const float* b1 = (const float*)d_in[4];
  const float* Wd = (const float*)d_in[5];
  const float* bd = (const float*)d_in[6];
  const int*   ei = (const int*)d_in[7];
  const int* e0 = ei;
  const int* e1 = ei + N_EDGES;
  float* out = (float*)d_out;

  const size_t NC = (size_t)N_NODES * CH;
  float* T0 = (float*)d_ws;    // current tangent (GEMM A)
  float* MM = T0 + NC;         // GEMM output
  float* T2 = MM + NC;         // tangent for aggregation
  float* S  = T2 + NC;         // aggregated tangent
  float* UB = S + NC;          // bias tangent (128)

  dim3 nodeGrid((N_NODES + 7) / 8), nodeBlk(256);
  const int mmBlocks = (MTILES + 7) / 8;   // 8 M-tiles (waves) per block

  hgcn_embed<<<nodeGrid, nodeBlk, 0, stream>>>(x, T0);

  const float* Ws[2]   = {W0, W1};
  const float* bs[2]   = {b0, b1};
  const int    ldws[2] = {CH + 1, CH};   // W0 is (128,129); skip its col 0
  const int    woff[2] = {1, 0};

  for (int l = 0; l < 2; ++l) {
    hgcn_bias<<<1, 32, 0, stream>>>(bs[l], UB);
    // each wave: 16 x 64 stripe; weight panel staged in LDS per block
    hgcn_wmma_mm<4><<<dim3(mmBlocks, CH / 64), 256, 0, stream>>>(
        T0, Ws[l], ldws[l], woff[l], nullptr, MM, CH, MTILES);
    hgcn_mobius<<<nodeGrid, nodeBlk, 0, stream>>>(MM, UB, T2);
    hgcn_zero<<<(int)(NC / 4 / 256), 256, 0, stream>>>(S, (int)(NC / 4));
    hgcn_scatter<<<N_EDGES * 32 / 256, 256, 0, stream>>>(e0, e1, T2, S);
    hgcn_actnext<<<nodeGrid, nodeBlk, 0, stream>>>(S, T0);
  }

  // head: out = t_final @ Wd^T + bd   (10000 x 128 x 16)
  hgcn_wmma_mm<1><<<dim3(mmBlocks, 1), 256, 0, stream>>>(
      T0, Wd, CH, 0, bd, out, 16, MTILES);
}